// RandomShiftsAug_56341380989251
// MI455X (gfx1250) — compile-verified
//
#include <hip/hip_runtime.h>
#include <stdint.h>

// RandomShiftsAug for MI455X (gfx1250).
// Pure gather/copy: out[p,i,j] = x[p, clamp(i+sy-4,0,83), clamp(j+sx-4,0,83)].
// Bandwidth-bound (~347 MB total -> ~15us floor at 23.3 TB/s). Uses the CDNA5
// async global<->LDS DMA path (ASYNCcnt): b32 gather loads into LDS, then
// wide b128 async stores from LDS to the contiguous output.

namespace {
constexpr int kNImg  = 2048;            // N * V
constexpr int kC     = 3;
constexpr int kH     = 84;
constexpr int kPad   = 4;               // 84 / 21
constexpr int kPlane = kH * kH;         // 7056
constexpr int kPlanes = kNImg * kC;     // 6144
constexpr int kTotal = kPlanes * kPlane;     // 43,352,064
constexpr int kSlots = 16;              // async loads in flight per thread
constexpr int kBlock = 256;             // 8 waves (wave32)
constexpr int kChunk = kSlots * kBlock; // 4096 elements per block
static_assert(kTotal % kChunk == 0, "exact tiling, no tail");
static_assert(kChunk < kPlane, "chunk spans at most two planes");
}

#if defined(__has_builtin)
#if __has_builtin(__builtin_amdgcn_global_load_async_to_lds_b32) && \
    __has_builtin(__builtin_amdgcn_global_store_async_from_lds_b128)
#define HAVE_ASYNC_LDS 1
#endif
#endif
#ifndef HAVE_ASYNC_LDS
#define HAVE_ASYNC_LDS 0
#endif

typedef int v4i __attribute__((ext_vector_type(4)));
typedef __attribute__((address_space(1))) int GASInt;  // global, b32
typedef __attribute__((address_space(3))) int LASInt;  // LDS, b32
typedef __attribute__((address_space(1))) v4i GASV4;   // global, b128
typedef __attribute__((address_space(3))) v4i LASV4;   // LDS, b128

__global__ __launch_bounds__(kBlock) void RandomShiftsAug_56341380989251_kernel(
    const float* __restrict__ x, const int* __restrict__ shift,
    float* __restrict__ out)
{
    const unsigned tid  = threadIdx.x;
    const unsigned base = (unsigned)blockIdx.x * (unsigned)kChunk;

    // ---- Uniform (per-block) precomputation: at most 2 planes / 2 images ----
    const unsigned p0   = base / (unsigned)kPlane;          // first plane
    const unsigned rem0 = base - p0 * (unsigned)kPlane;     // offset into it
    const unsigned pb0  = p0 * (unsigned)kPlane;            // its element base
    const unsigned p1   = (p0 + 1u > (unsigned)(kPlanes - 1))
                            ? (unsigned)(kPlanes - 1) : p0 + 1u;  // clamped (avoid OOB shift read)
    const unsigned m0   = p0 / (unsigned)kC;
    const unsigned m1   = p1 / (unsigned)kC;
    const int sx0 = shift[2u * m0 + 0u], sy0 = shift[2u * m0 + 1u];
    const int sx1 = shift[2u * m1 + 0u], sy1 = shift[2u * m1 + 1u];

#if HAVE_ASYNC_LDS
    __shared__ float lds[kChunk];  // 16 KB staging, linear by chunk offset o

    // Phase 1: 16 async b32 gather-loads (global -> LDS); per-lane clamped
    // addresses implement the replicate padding exactly.
#pragma unroll
    for (int k = 0; k < kSlots; ++k) {
        unsigned o  = (unsigned)(k * kBlock) + tid;      // 0..4095
        unsigned re = rem0 + o;
        unsigned nx = (re >= (unsigned)kPlane) ? 1u : 0u;
        unsigned r  = nx ? re - (unsigned)kPlane : re;   // offset within plane
        unsigned i  = r / (unsigned)kH;                  // row (one small div)
        unsigned j  = r - i * (unsigned)kH;              // col
        int sx = nx ? sx1 : sx0;
        int sy = nx ? sy1 : sy0;
        int si = (int)i + sy - kPad;
        si = si < 0 ? 0 : (si > kH - 1 ? kH - 1 : si);
        int sj = (int)j + sx - kPad;
        sj = sj < 0 ? 0 : (sj > kH - 1 ? kH - 1 : sj);
        unsigned src = (nx ? pb0 + (unsigned)kPlane : pb0)
                     + (unsigned)si * (unsigned)kH + (unsigned)sj;
        GASInt* gp = (GASInt*)(uintptr_t)(x + src);
        LASInt* lp = (LASInt*)(unsigned)(uintptr_t)(&lds[o]);
        __builtin_amdgcn_global_load_async_to_lds_b32(gp, lp, /*offset=*/0, /*cpol=*/0);
    }

    // Each wave drains its own async loads, then block-wide barrier so every
    // thread may store data loaded by any wave.
    __builtin_amdgcn_s_wait_asynccnt(0);
    __syncthreads();

    // Phase 2: 4 async b128 stores (LDS -> global), contiguous & 16B aligned.
#pragma unroll
    for (int s = 0; s < 4; ++s) {
        unsigned o = (unsigned)(s * (kChunk / 4)) + tid * 4u;  // multiple of 4
        GASV4* gp = (GASV4*)(uintptr_t)(out + base + o);
        LASV4* lp = (LASV4*)(unsigned)(uintptr_t)(&lds[o]);
        __builtin_amdgcn_global_store_async_from_lds_b128(gp, lp, /*offset=*/0, /*cpol=*/0);
    }
    // S_ENDPGM performs an implicit wait-idle, draining outstanding async stores.
#else
    // Fallback: direct coalesced register gather-copy (still bandwidth-bound).
#pragma unroll
    for (int k = 0; k < kSlots; ++k) {
        unsigned o  = (unsigned)(k * kBlock) + tid;
        unsigned re = rem0 + o;
        unsigned nx = (re >= (unsigned)kPlane) ? 1u : 0u;
        unsigned r  = nx ? re - (unsigned)kPlane : re;
        unsigned i  = r / (unsigned)kH;
        unsigned j  = r - i * (unsigned)kH;
        int sx = nx ? sx1 : sx0;
        int sy = nx ? sy1 : sy0;
        int si = (int)i + sy - kPad;
        si = si < 0 ? 0 : (si > kH - 1 ? kH - 1 : si);
        int sj = (int)j + sx - kPad;
        sj = sj < 0 ? 0 : (sj > kH - 1 ? kH - 1 : sj);
        unsigned src = (nx ? pb0 + (unsigned)kPlane : pb0)
                     + (unsigned)si * (unsigned)kH + (unsigned)sj;
        out[base + o] = x[src];
    }
#endif
}

extern "C" void kernel_launch(void* const* d_in, const int* in_sizes, int n_in,
                              void* d_out, int out_size, void* d_ws, size_t ws_size,
                              hipStream_t stream) {
    (void)in_sizes; (void)n_in; (void)out_size; (void)d_ws; (void)ws_size;
    const float* x     = (const float*)d_in[0];
    const int*   shift = (const int*)d_in[1];
    float*       out   = (float*)d_out;

    dim3 grid(kTotal / kChunk);  // 10,584 blocks
    dim3 block(kBlock);          // 256 threads = 8 wave32
    hipLaunchKernelGGL(RandomShiftsAug_56341380989251_kernel, grid, block, 0, stream,
                       x, shift, out);
}